// GraphAttentionLayer_37469294690432
// MI455X (gfx1250) — compile-verified
//
#include <hip/hip_runtime.h>
#include <hip/hip_bf16.h>

typedef __attribute__((ext_vector_type(2))) float v2f;
typedef __attribute__((ext_vector_type(8))) float v8f;

#define FEAT 128

// ---------------------------------------------------------------------------
// Kernel 1: per-node scores e[n] = dot(x[n,:], a) via V_WMMA_F32_16X16X4_F32.
// One wave of 32 lanes computes 16 nodes; K=128 accumulated in 32 steps of 4.
// Tail handling: clamp the node index instead of predicating the load —
// D row m depends only on A row m, and out-of-range stores are guarded, so
// clamped (duplicate) rows are harmless. This keeps EXEC all-1s and the loop
// body branch-free (pure load -> wmma).
// ---------------------------------------------------------------------------
__global__ void gat_scores_wmma(const float* __restrict__ x,
                                const float* __restrict__ a,
                                float* __restrict__ e,
                                int n_nodes) {
    const int wave = threadIdx.x >> 5;
    const int lane = threadIdx.x & 31;
    const int wavesPerBlock = blockDim.x >> 5;
    const int base = (blockIdx.x * wavesPerBlock + wave) * 16;
    if (base >= n_nodes) return;               // wave-uniform exit: EXEC stays all-1s

    const int half = lane >> 4;                // 0: K={0,1}, 1: K={2,3} within chunk
    const int mrow = lane & 15;                // A-matrix row M
    int node = base + mrow;
    if (node >= n_nodes) node = n_nodes - 1;   // clamp: branch-free tail

    const float* __restrict__ xrow = x + (size_t)node * FEAT + half * 2;
    const float* __restrict__ brow = a + half * 2;

    v8f c = {};
#pragma unroll
    for (int k0 = 0; k0 < FEAT; k0 += 4) {
        const float2 xv = *(const float2*)(xrow + k0);
        const float2 bv = *(const float2*)(brow + k0);   // B[k][n] = a[k] for all n
        v2f av; av.x = xv.x; av.y = xv.y;
        v2f bf; bf.x = bv.x; bf.y = bv.y;
        // (neg_a, A, neg_b, B, c_mod, C, reuse_a, reuse_b)
        c = __builtin_amdgcn_wmma_f32_16x16x4_f32(false, av, false, bf,
                                                  (short)0, c, false, false);
    }

    // C/D layout: VGPR v -> M=v (lanes 0-15) / M=8+v (lanes 16-31), N = lane%16.
    if (mrow == 0) {                           // N == 0 column holds the score
#pragma unroll
        for (int v = 0; v < 8; ++v) {
            const int nd = base + half * 8 + v;
            if (nd < n_nodes) e[nd] = c[v];
        }
    }
}

// ---------------------------------------------------------------------------
// Kernel 2: init segment-max to -inf, denom to 0 (re-run every call).
// ---------------------------------------------------------------------------
__global__ void gat_init_nodes(float* __restrict__ m, float* __restrict__ denom, int n) {
    const int i = blockIdx.x * blockDim.x + threadIdx.x;
    if (i < n) {
        m[i] = -__builtin_inff();
        denom[i] = 0.0f;
    }
}

// Float atomic-max via signed-int max (vals >= 0) / unsigned-int min (vals < 0).
__device__ inline void atomicMaxF(float* addr, float val) {
    if (val >= 0.0f)
        atomicMax((int*)addr, __float_as_int(val));
    else
        atomicMin((unsigned int*)addr, __float_as_uint(val));
}

// ---------------------------------------------------------------------------
// Kernel 3: segment max over rows of gathered destination scores.
// ---------------------------------------------------------------------------
__global__ void gat_edge_max(const int* __restrict__ row, const int* __restrict__ col,
                             const float* __restrict__ e, float* __restrict__ m, int nE) {
    const int k = blockIdx.x * blockDim.x + threadIdx.x;
    if (k < nE) atomicMaxF(&m[row[k]], e[col[k]]);
}

// ---------------------------------------------------------------------------
// Kernel 4: ex = exp(vals - m[row]); denom = segment_sum(ex).
// unsafeAtomicAdd -> global_atomic_add_f32 (hardware float atomic, no CAS loop).
// ---------------------------------------------------------------------------
__global__ void gat_edge_expsum(const int* __restrict__ row, const int* __restrict__ col,
                                const float* __restrict__ e, const float* __restrict__ m,
                                float* __restrict__ ex, float* __restrict__ denom, int nE) {
    const int k = blockIdx.x * blockDim.x + threadIdx.x;
    if (k < nE) {
        const float v = expf(e[col[k]] - m[row[k]]);
        ex[k] = v;
        unsafeAtomicAdd(&denom[row[k]], v);
    }
}

// ---------------------------------------------------------------------------
// Kernel 5: h[row] += (ex/denom[row]) * x[col].  One wave per edge:
// 32 lanes x float4 = full 128-feature row, coalesced 512B load of x[col].
// ---------------------------------------------------------------------------
__global__ void gat_edge_aggregate(const int* __restrict__ row, const int* __restrict__ col,
                                   const float* __restrict__ x,
                                   const float* __restrict__ ex, const float* __restrict__ denom,
                                   float* __restrict__ h, int nE) {
    const int wave = threadIdx.x >> 5;
    const int lane = threadIdx.x & 31;
    const int k = blockIdx.x * (blockDim.x >> 5) + wave;
    if (k >= nE) return;

    const int r = row[k];
    const int cn = col[k];
    const float att = ex[k] / denom[r];

    const float4 xv = *(const float4*)(x + (size_t)cn * FEAT + lane * 4);
    float* hp = h + (size_t)r * FEAT + lane * 4;
    unsafeAtomicAdd(hp + 0, att * xv.x);
    unsafeAtomicAdd(hp + 1, att * xv.y);
    unsafeAtomicAdd(hp + 2, att * xv.z);
    unsafeAtomicAdd(hp + 3, att * xv.w);
}

// ---------------------------------------------------------------------------
extern "C" void kernel_launch(void* const* d_in, const int* in_sizes, int n_in,
                              void* d_out, int out_size, void* d_ws, size_t ws_size,
                              hipStream_t stream) {
    const float* x   = (const float*)d_in[0];
    const float* a   = (const float*)d_in[1];
    const int*   row = (const int*)d_in[2];
    const int*   col = (const int*)d_in[3];
    float*       h   = (float*)d_out;

    const int nNodes = in_sizes[0] / FEAT;   // 50000
    const int nE     = in_sizes[2];          // 800000

    // Workspace layout (floats): e[nNodes] | m[nNodes] | denom[nNodes] | ex[nE]
    float* e     = (float*)d_ws;
    float* m     = e + nNodes;
    float* denom = m + nNodes;
    float* ex    = denom + nNodes;

    // Output accumulators must be zero every call.
    hipMemsetAsync(d_out, 0, (size_t)out_size * sizeof(float), stream);

    // 1) scores via WMMA: 8 waves/block, 16 nodes/wave -> 128 nodes/block
    {
        const int nodesPerBlock = 8 * 16;
        const int grid = (nNodes + nodesPerBlock - 1) / nodesPerBlock;
        gat_scores_wmma<<<grid, 256, 0, stream>>>(x, a, e, nNodes);
    }
    // 2) init m/denom
    gat_init_nodes<<<(nNodes + 255) / 256, 256, 0, stream>>>(m, denom, nNodes);
    // 3) segment max
    gat_edge_max<<<(nE + 255) / 256, 256, 0, stream>>>(row, col, e, m, nE);
    // 4) exp + segment sum
    gat_edge_expsum<<<(nE + 255) / 256, 256, 0, stream>>>(row, col, e, m, ex, denom, nE);
    // 5) normalize + scatter-accumulate (1 wave / edge, 8 edges / block)
    {
        const int edgesPerBlock = 256 / 32;
        const int grid = (nE + edgesPerBlock - 1) / edgesPerBlock;
        gat_edge_aggregate<<<grid, 256, 0, stream>>>(row, col, x, ex, denom, h, nE);
    }
}